// SAGE_884763263088
// MI455X (gfx1250) — compile-verified
//
#include <hip/hip_runtime.h>
#include <stdint.h>

typedef float v2f __attribute__((ext_vector_type(2)));
typedef float v8f __attribute__((ext_vector_type(8)));

#define NNODES 100000
#define NEDGES 600000
#define FIN 128
#define FHID 256
#define FOUT 64

// Monotone float -> u32 map: unsigned compare order == float order.
// enc(-inf) = 0x007FFFFF (minimum over all non-NaN floats).
#define ENC_NEG_INF 0x007FFFFFu
__device__ __forceinline__ unsigned enc_f32(float x) {
    unsigned u = __float_as_uint(x);
    return (u & 0x80000000u) ? ~u : (u | 0x80000000u);
}
// Inverse map; untouched sentinel (-inf, i.e. node with no in-edges) -> 0.0f
__device__ __forceinline__ float dec_f32(unsigned u) {
    float f = (u & 0x80000000u) ? __uint_as_float(u & 0x7FFFFFFFu)
                                : __uint_as_float(~u);
    return (u == ENC_NEG_INF) ? 0.0f : f;
}

__global__ void fill_u32x4(uint4* __restrict__ p, int n4, unsigned val) {
    int i = blockIdx.x * blockDim.x + threadIdx.x;
    if (i < n4) p[i] = make_uint4(val, val, val, val);
}

// One wave per edge: all 32 lanes read the (coalesced) source feature row as
// float4s from L2-resident features, encode, and scatter via native u32
// atomic-max (non-returning -> GLOBAL_ATOMIC_MAX_U32, STOREcnt path).
template <int F>
__global__ void scatter_max(const float* __restrict__ feat,
                            const long long* __restrict__ src,
                            const long long* __restrict__ dst,
                            unsigned* __restrict__ agg, int nedges) {
    int gtid = blockIdx.x * blockDim.x + threadIdx.x;
    int e    = gtid >> 5;
    int lane = gtid & 31;
    if (e >= nedges) return;
    int s = (int)src[e];
    int d = (int)dst[e];
    const float4* frow = (const float4*)(feat + (size_t)s * F);
    unsigned*     arow = agg + (size_t)d * F;
#pragma unroll
    for (int c = lane; c < F / 4; c += 32) {
        float4 v = frow[c];
        atomicMax(arow + 4 * c + 0, enc_f32(v.x));
        atomicMax(arow + 4 * c + 1, enc_f32(v.y));
        atomicMax(arow + 4 * c + 2, enc_f32(v.z));
        atomicMax(arow + 4 * c + 3, enc_f32(v.w));
    }
}

// C = dec(A1enc)@B1 + A2@B2 + bias  (optionally relu * mask), f32 WMMA 16x16x4.
// A1 is the encoded-u32 aggregation buffer; decode fused into the frag load.
// Each wave: 16 rows x 32 cols (two 16x16 tiles / two accumulator chains).
// A frag (ISA 7.12.2, 32-bit A 16x4): lane m = row, half-wave kh selects K pair.
// B frag (4x16): lane m = col, VGPR v -> K row = kh*2 + v.
// C/D: element v of lane l -> row = v + (l>>4)*8, col = l&15.
template <int K, int NCOL, bool RELU_MASK>
__global__ void sage_gemm(const unsigned* __restrict__ A1enc, const float* __restrict__ B1,
                          const float* __restrict__ A2, const float* __restrict__ B2,
                          const float* __restrict__ bias, const float* __restrict__ mask,
                          float* __restrict__ C, int nrows) {
    const int WPS = NCOL / 32;  // waves per 16-row slab
    int gwave    = blockIdx.x * (blockDim.x >> 5) + (threadIdx.x >> 5);
    int row_tile = gwave / WPS;
    int colbase  = (gwave % WPS) * 32;
    if (row_tile * 16 >= nrows) return;

    int lane = threadIdx.x & 31;
    int m    = lane & 15;
    int kh   = lane >> 4;
    int rowA = row_tile * 16 + m;

    v8f c0 = {0.f, 0.f, 0.f, 0.f, 0.f, 0.f, 0.f, 0.f};
    v8f c1 = c0;

    const uint2* a1p = (const uint2*)(A1enc + (size_t)rowA * K + kh * 2);
    const float* a2p = A2 + (size_t)rowA * K + kh * 2;
    const float* b1p = B1 + (size_t)(kh * 2) * NCOL + colbase + m;
    const float* b2p = B2 + (size_t)(kh * 2) * NCOL + colbase + m;

#pragma unroll 4
    for (int k = 0; k < K; k += 4) {
        uint2 ue = a1p[k / 2];  // elements k+kh*2, k+kh*2+1 (uint2 units)
        v2f a;  a.x  = dec_f32(ue.x);     a.y  = dec_f32(ue.y);
        v2f b0; b0.x = b1p[(size_t)k * NCOL];      b0.y = b1p[(size_t)(k + 1) * NCOL];
        v2f b1; b1.x = b1p[(size_t)k * NCOL + 16]; b1.y = b1p[(size_t)(k + 1) * NCOL + 16];
        c0 = __builtin_amdgcn_wmma_f32_16x16x4_f32(false, a, false, b0, (short)0, c0, false, false);
        c1 = __builtin_amdgcn_wmma_f32_16x16x4_f32(false, a, false, b1, (short)0, c1, false, false);
    }
#pragma unroll 4
    for (int k = 0; k < K; k += 4) {
        v2f a;  a.x  = a2p[k];            a.y  = a2p[k + 1];
        v2f b0; b0.x = b2p[(size_t)k * NCOL];      b0.y = b2p[(size_t)(k + 1) * NCOL];
        v2f b1; b1.x = b2p[(size_t)k * NCOL + 16]; b1.y = b2p[(size_t)(k + 1) * NCOL + 16];
        c0 = __builtin_amdgcn_wmma_f32_16x16x4_f32(false, a, false, b0, (short)0, c0, false, false);
        c1 = __builtin_amdgcn_wmma_f32_16x16x4_f32(false, a, false, b1, (short)0, c1, false, false);
    }

    int   col0 = colbase + m;
    float bia0 = bias[col0];
    float bia1 = bias[col0 + 16];
#pragma unroll
    for (int v = 0; v < 8; v++) {
        int    row = row_tile * 16 + kh * 8 + v;
        size_t off = (size_t)row * NCOL + col0;
        float  x0  = c0[v] + bia0;
        float  x1  = c1[v] + bia1;
        if (RELU_MASK) {
            x0 = fmaxf(x0, 0.0f) * mask[off];
            x1 = fmaxf(x1, 0.0f) * mask[off + 16];
        }
        C[off]      = x0;
        C[off + 16] = x1;
    }
}

extern "C" void kernel_launch(void* const* d_in, const int* in_sizes, int n_in,
                              void* d_out, int out_size, void* d_ws, size_t ws_size,
                              hipStream_t stream) {
    const float*     x    = (const float*)d_in[0];
    const long long* ei   = (const long long*)d_in[1];  // int64 [2, E]
    const float*     W_l1 = (const float*)d_in[2];
    const float*     b_l1 = (const float*)d_in[3];
    const float*     W_r1 = (const float*)d_in[4];
    const float*     W_l2 = (const float*)d_in[5];
    const float*     b_l2 = (const float*)d_in[6];
    const float*     W_r2 = (const float*)d_in[7];
    const float*     mask = (const float*)d_in[8];
    const long long* src  = ei;
    const long long* dst  = ei + NEDGES;

    char*     ws   = (char*)d_ws;
    unsigned* agg1 = (unsigned*)ws;                                              // N*128 u32
    float*    h    = (float*)(ws + (size_t)NNODES * FIN * 4);                    // N*256 f32
    unsigned* agg2 = (unsigned*)(ws + (size_t)NNODES * (FIN + FHID) * 4);        // N*256 u32
    float*    out  = (float*)d_out;

    const int  n1       = NNODES * FIN;   // 12.8M
    const int  n2       = NNODES * FHID;  // 25.6M
    const long sthreads = (long)NEDGES * 32;

    // ---- layer 1 ----
    fill_u32x4<<<(n1 / 4 + 255) / 256, 256, 0, stream>>>((uint4*)agg1, n1 / 4, ENC_NEG_INF);
    scatter_max<FIN><<<(unsigned)((sthreads + 255) / 256), 256, 0, stream>>>(x, src, dst, agg1, NEDGES);
    // h = relu(dec(agg1)@W_l1 + x@W_r1 + b_l1) * mask  (one 256-thread block per 16-row slab)
    sage_gemm<FIN, FHID, true><<<NNODES / 16, 256, 0, stream>>>(
        agg1, W_l1, x, W_r1, b_l1, mask, h, NNODES);

    // ---- layer 2 ----
    fill_u32x4<<<(n2 / 4 + 255) / 256, 256, 0, stream>>>((uint4*)agg2, n2 / 4, ENC_NEG_INF);
    scatter_max<FHID><<<(unsigned)((sthreads + 255) / 256), 256, 0, stream>>>(h, src, dst, agg2, NEDGES);
    // out = dec(agg2)@W_l2 + h@W_r2 + b_l2
    const int waves2 = (NNODES / 16) * (FOUT / 32);  // 12500
    sage_gemm<FHID, FOUT, false><<<(waves2 + 7) / 8, 256, 0, stream>>>(
        agg2, W_l2, h, W_r2, b_l2, nullptr, out, NNODES);
}